// Generator3DLUT_zero_86122684220303
// MI455X (gfx1250) — compile-verified
//
#include <hip/hip_runtime.h>

// 3D LUT trilinear apply: x (4,3,1080,1920) f32, LUT (3,33,33,33) f32 -> out (4,3,1080,1920) f32
// Memory-bound: ~199 MB HBM traffic -> ~8.6us roofline @ 23.3 TB/s. LUT (575 KB AoS) lives in L2.
// WMMA inapplicable: per-pixel gather, no shared matmul operand; intensity ~2 FLOP/B.

#define LUT_DIM   33
#define LUT_N     (LUT_DIM * LUT_DIM * LUT_DIM)   // 35937
#define S_G       (LUT_DIM)                       // g stride = 33
#define S_B       (LUT_DIM * LUT_DIM)             // b stride = 1089
#define PLANE     (1080 * 1920)                   // 2073600
#define NBATCH    4
#define NPIX      (NBATCH * PLANE)                // 8294400
#define VEC       4
#define NTHREADS  (NPIX / VEC)                    // 2073600 = 8100 * 256

typedef float v4f __attribute__((ext_vector_type(4)));
typedef float v3f __attribute__((ext_vector_type(3)));

__device__ __forceinline__ v3f lerp3(v3f a, v3f b, float t) {
    return a + t * (b - a);   // 3x (v_sub + v_fma), no work on the pad lane
}

__device__ __forceinline__ void idxfrac(float v, int& idx, float& frac) {
    float s = fminf(fmaxf(v, 0.0f), 1.0f) * (float)(LUT_DIM - 1);
    int i = (int)s;                       // s >= 0, so cast == floor
    i = (i > LUT_DIM - 2) ? (LUT_DIM - 2) : i;
    idx  = i;
    frac = s - (float)i;                  // ==1.0 at the clamped top edge (matches ref)
}

// ---------------------------------------------------------------------------
// Kernel 1: planar (3,33,33,33) -> AoS float4[33^3] in workspace
// ---------------------------------------------------------------------------
__global__ __launch_bounds__(256) void lut_transpose_kernel(
        const float* __restrict__ lut, v4f* __restrict__ lut4) {
    int e = blockIdx.x * 256 + threadIdx.x;
    if (e < LUT_N) {
        v4f v;
        v.x = lut[e];
        v.y = lut[LUT_N + e];
        v.z = lut[2 * LUT_N + e];
        v.w = 0.0f;
        lut4[e] = v;
    }
}

// ---------------------------------------------------------------------------
// Kernel 2 (fast path): AoS LUT, 4 px/thread, b128 gathers, NT image traffic
// ---------------------------------------------------------------------------
__global__ __launch_bounds__(256) void lut_apply_aos_kernel(
        const float* __restrict__ x, const v4f* __restrict__ lut4,
        float* __restrict__ out) {
    const int tid = blockIdx.x * 256 + threadIdx.x;

    // Warm the LUT into cache: 35937 float4 = 4493 x 128B lines -> global_prefetch_b8
    if (tid < (LUT_N + 7) / 8) {
        __builtin_prefetch((const void*)(lut4 + (size_t)tid * 8), 0, 3);
    }

    const int n = tid / (PLANE / VEC);
    const int p = tid % (PLANE / VEC);
    const size_t base = (size_t)n * 3 * PLANE + (size_t)p * VEC;

    // Streaming image data: non-temporal so it never competes with the LUT in cache.
    v4f rv = __builtin_nontemporal_load((const v4f*)(x + base));
    v4f gv = __builtin_nontemporal_load((const v4f*)(x + base + PLANE));
    v4f bv = __builtin_nontemporal_load((const v4f*)(x + base + 2 * PLANE));

    v4f outr, outg, outb;

    #pragma unroll
    for (int k = 0; k < VEC; ++k) {
        int ir, ig, ib;
        float fr, fg, fb;
        idxfrac(rv[k], ir, fr);
        idxfrac(gv[k], ig, fg);
        idxfrac(bv[k], ib, fb);

        const v4f* c = lut4 + (ib * S_B + ig * S_G + ir);
        // 8 corners: one shifted voffset + immediate offsets; b128 each
        v4f c000 = c[0];
        v4f c001 = c[1];
        v4f c010 = c[S_G];
        v4f c011 = c[S_G + 1];
        v4f c100 = c[S_B];
        v4f c101 = c[S_B + 1];
        v4f c110 = c[S_B + S_G];
        v4f c111 = c[S_B + S_G + 1];

        // interpolate only the 3 live components
        v3f a00 = lerp3(c000.xyz, c001.xyz, fr);
        v3f a01 = lerp3(c010.xyz, c011.xyz, fr);
        v3f a10 = lerp3(c100.xyz, c101.xyz, fr);
        v3f a11 = lerp3(c110.xyz, c111.xyz, fr);
        v3f b0  = lerp3(a00, a01, fg);
        v3f b1  = lerp3(a10, a11, fg);
        v3f res = lerp3(b0, b1, fb);

        outr[k] = res.x;
        outg[k] = res.y;
        outb[k] = res.z;
    }

    __builtin_nontemporal_store(outr, (v4f*)(out + base));
    __builtin_nontemporal_store(outg, (v4f*)(out + base + PLANE));
    __builtin_nontemporal_store(outb, (v4f*)(out + base + 2 * PLANE));
}

// ---------------------------------------------------------------------------
// Fallback (no workspace): gathers straight from the planar LUT
// ---------------------------------------------------------------------------
__global__ __launch_bounds__(256) void lut_apply_planar_kernel(
        const float* __restrict__ x, const float* __restrict__ lut,
        float* __restrict__ out) {
    const int tid = blockIdx.x * 256 + threadIdx.x;

    if (tid < (3 * LUT_N + 31) / 32) {
        __builtin_prefetch((const void*)(lut + (size_t)tid * 32), 0, 3);
    }

    const int n = tid / (PLANE / VEC);
    const int p = tid % (PLANE / VEC);
    const size_t base = (size_t)n * 3 * PLANE + (size_t)p * VEC;

    v4f rv = __builtin_nontemporal_load((const v4f*)(x + base));
    v4f gv = __builtin_nontemporal_load((const v4f*)(x + base + PLANE));
    v4f bv = __builtin_nontemporal_load((const v4f*)(x + base + 2 * PLANE));

    v4f och[3];

    #pragma unroll
    for (int k = 0; k < VEC; ++k) {
        int ir, ig, ib;
        float fr, fg, fb;
        idxfrac(rv[k], ir, fr);
        idxfrac(gv[k], ig, fg);
        idxfrac(bv[k], ib, fb);

        const int cb = ib * S_B + ig * S_G + ir;
        #pragma unroll
        for (int ch = 0; ch < 3; ++ch) {
            const float* L = lut + ch * LUT_N + cb;
            float a00 = L[0]         + fr * (L[1]             - L[0]);
            float a01 = L[S_G]       + fr * (L[S_G + 1]       - L[S_G]);
            float a10 = L[S_B]       + fr * (L[S_B + 1]       - L[S_B]);
            float a11 = L[S_B + S_G] + fr * (L[S_B + S_G + 1] - L[S_B + S_G]);
            float b0  = a00 + fg * (a01 - a00);
            float b1  = a10 + fg * (a11 - a10);
            och[ch][k] = b0 + fb * (b1 - b0);
        }
    }

    __builtin_nontemporal_store(och[0], (v4f*)(out + base));
    __builtin_nontemporal_store(och[1], (v4f*)(out + base + PLANE));
    __builtin_nontemporal_store(och[2], (v4f*)(out + base + 2 * PLANE));
}

// ---------------------------------------------------------------------------
extern "C" void kernel_launch(void* const* d_in, const int* in_sizes, int n_in,
                              void* d_out, int out_size, void* d_ws, size_t ws_size,
                              hipStream_t stream) {
    const float* x   = (const float*)d_in[0];   // (4,3,1080,1920) f32
    const float* lut = (const float*)d_in[1];   // (3,33,33,33)    f32
    float* out = (float*)d_out;

    const size_t ws_needed = (size_t)LUT_N * sizeof(v4f);  // 574,992 B

    if (d_ws != nullptr && ws_size >= ws_needed) {
        v4f* lut4 = (v4f*)d_ws;
        lut_transpose_kernel<<<(LUT_N + 255) / 256, 256, 0, stream>>>(lut, lut4);
        lut_apply_aos_kernel<<<NTHREADS / 256, 256, 0, stream>>>(x, lut4, out);
    } else {
        lut_apply_planar_kernel<<<NTHREADS / 256, 256, 0, stream>>>(x, lut, out);
    }
}